// Clahe_188978561390
// MI455X (gfx1250) — compile-verified
//
#include <hip/hip_runtime.h>

// CLAHE for (8,3,1024,1024) f32, GRID=8x8, 256 bins, clip=40.
// Pass 1: one block per tile -> LDS histogram (ds_add atomics), clip +
//         redistribute, then ONE WAVE computes the 256-bin CDF as two
//         triangular 16x16x16 matmuls on V_WMMA_F32_16X16X4_F32 (exact f32
//         integer arithmetic), and writes the 256-entry LUT.
// Pass 2: one block per (image, 16-row band) -> stage the two needed
//         8x256 LUT rows into LDS via GLOBAL_LOAD_ASYNC_TO_LDS_B128
//         (ASYNCcnt path), bilinear-blend per pixel.

#define BCN   24            // B*C
#define Hh    1024
#define Ww    1024
#define THT   128           // tile height
#define TWT   128           // tile width
#define NB    256
#define PIXELS (THT * TWT)  // 16384
#define CLIPV 2560.0f       // floor(40*16384/256)

typedef float v2f __attribute__((ext_vector_type(2)));
typedef float v8f __attribute__((ext_vector_type(8)));
// Exact parameter type reported by clang for the async-LDS builtins:
typedef int v4ig __attribute__((vector_size(4 * sizeof(int))));
typedef __attribute__((address_space(1))) v4ig* as1_v4ip;
typedef __attribute__((address_space(3))) v4ig* as3_v4ip;

__global__ __launch_bounds__(256)
void clahe_lut_kernel(const float* __restrict__ img, float* __restrict__ lut)
{
    __shared__ unsigned int histU[NB];
    __shared__ float histF[NB];
    __shared__ float red[NB];
    __shared__ float rowtot[16];
    __shared__ float bcast[2];

    const int t    = threadIdx.x;
    const int tile = blockIdx.x;          // bc*64 + gy*8 + gx
    const int gx   = tile & 7;
    const int gy   = (tile >> 3) & 7;
    const int bc   = tile >> 6;

    histU[t] = 0u;
    __syncthreads();

    // ---- per-tile histogram via LDS atomics (coalesced 128-wide rows) ----
    const long tileBase = (long)bc * (Hh * Ww) + (long)(gy * THT) * Ww + (long)(gx * TWT);
    for (int i = 0; i < PIXELS / 256; ++i) {
        int p = t + i * 256;
        int r = p >> 7;            // / TWT
        int c = p & (TWT - 1);
        float v = img[tileBase + (long)r * Ww + c];
        v = fminf(fmaxf(v, 0.0f), 1.0f);
        int bin = (int)(v * 256.0f);
        bin = bin > 255 ? 255 : bin;
        atomicAdd(&histU[bin], 1u);
    }
    __syncthreads();

    // ---- clip + redistribute (exact f32 integer arithmetic) ----
    float hc = fminf((float)histU[t], CLIPV);
    histF[t] = hc;
    red[t]   = hc;
    __syncthreads();
    for (int s = 128; s > 0; s >>= 1) {
        if (t < s) red[t] += red[t + s];
        __syncthreads();
    }
    if (t == 0) {
        float clipped  = (float)PIXELS - red[0];
        float redist   = floorf(clipped / 256.0f);
        float residual = clipped - redist * 256.0f;
        bcast[0] = redist;
        bcast[1] = residual;
    }
    __syncthreads();
    histF[t] = histF[t] + bcast[0] + (((float)t < bcast[1]) ? 1.0f : 0.0f);
    __syncthreads();

    // ---- CDF via WMMA: Hm(16x16), cdf = Hm*U + Lstrict*S  (wave 0 only) ----
    // A layout (16x4 f32): lanes 0-15 row=lane, VGPR0=K0 VGPR1=K1;
    //                      lanes 16-31 row=lane-16, VGPR0=K2 VGPR1=K3.
    // B layout (4x16 f32): VGPR v: lanes 0-15 (row=v, col=lane);
    //                      lanes 16-31 (row=v+2, col=lane-16).
    // C layout (16x16):    VGPR v: lanes 0-15 (M=v), lanes 16-31 (M=v+8), N=lane&15.
    if (t < 32) {
        const int m  = t & 15;     // A-row / B-col / C-col for this lane
        const int hi = t >> 4;

        v8f acc = {0.f, 0.f, 0.f, 0.f, 0.f, 0.f, 0.f, 0.f};

        // Stage 1: row-wise inclusive prefix:  acc = Hm x U  (U[k,n] = k<=n)
        #pragma unroll
        for (int kc = 0; kc < 4; ++kc) {
            const int k0 = 4 * kc + 2 * hi;
            v2f a, b;
            a.x = histF[16 * m + k0];
            a.y = histF[16 * m + k0 + 1];
            b.x = (k0     <= m) ? 1.0f : 0.0f;
            b.y = (k0 + 1 <= m) ? 1.0f : 0.0f;
            acc = __builtin_amdgcn_wmma_f32_16x16x4_f32(
                false, a, false, b, (short)0, acc, false, false);
        }

        // Row totals = last column (held by lanes with m==15)
        if (m == 15) {
            #pragma unroll
            for (int v = 0; v < 8; ++v) rowtot[v + 8 * hi] = acc[v];
        }
        asm volatile("s_wait_dscnt 0" ::: "memory");

        // Stage 2: acc += Lstrict x S, S[k,:] = rowtot[k]  (adds row offsets)
        #pragma unroll
        for (int kc = 0; kc < 4; ++kc) {
            const int k0 = 4 * kc + 2 * hi;
            v2f a, b;
            a.x = (k0     < m) ? 1.0f : 0.0f;
            a.y = (k0 + 1 < m) ? 1.0f : 0.0f;
            b.x = rowtot[k0];
            b.y = rowtot[k0 + 1];
            acc = __builtin_amdgcn_wmma_f32_16x16x4_f32(
                false, a, false, b, (short)0, acc, false, false);
        }

        // acc[v] = cdf(bin = 16*(v+8*hi) + m)  ->  LUT
        float* lt = lut + (long)tile * NB;
        #pragma unroll
        for (int v = 0; v < 8; ++v) {
            float l = floorf(acc[v] * (255.0f / 16384.0f));
            l = fminf(fmaxf(l, 0.0f), 255.0f) * (1.0f / 255.0f);
            lt[16 * (v + 8 * hi) + m] = l;
        }
    }
}

__global__ __launch_bounds__(256)
void clahe_apply_kernel(const float* __restrict__ img,
                        const float* __restrict__ lut,
                        float* __restrict__ out)
{
    __shared__ __align__(16) float lutA[8 * NB];   // tile-row y0, all 8 columns (8 KB)
    __shared__ __align__(16) float lutB[8 * NB];   // tile-row y1 (8 KB)

    const int t  = threadIdx.x;
    const int bc = blockIdx.y;
    const int yb = blockIdx.x * 16;  // 16-row band; y0 uniform within band

    float tyb = (yb + 0.5f) / 128.0f - 0.5f;
    tyb = fminf(fmaxf(tyb, 0.0f), 7.0f);
    const int y0 = (int)tyb;
    const int y1 = min(y0 + 1, 7);

    // ---- stage LUT rows y0/y1 into LDS via async memory->LDS (ASYNCcnt) ----
    // AS1 ptr = flat global address; AS3 ptr = low 32 bits of flat shared
    // address (== LDS byte offset per the gfx1250 aperture mapping).
    const float4* la4 = (const float4*)(lut + ((long)bc * 64 + y0 * 8) * NB);
    const float4* lb4 = (const float4*)(lut + ((long)bc * 64 + y1 * 8) * NB);
    float4* lutA4 = (float4*)lutA;   // 512 float4
    float4* lutB4 = (float4*)lutB;
    #pragma unroll
    for (int i = 0; i < 2; ++i) {
        __builtin_amdgcn_global_load_async_to_lds_b128(
            (as1_v4ip)(uintptr_t)(la4 + i * 256 + t),
            (as3_v4ip)(unsigned)(uintptr_t)(lutA4 + i * 256 + t),
            0, 0);
        __builtin_amdgcn_global_load_async_to_lds_b128(
            (as1_v4ip)(uintptr_t)(lb4 + i * 256 + t),
            (as3_v4ip)(unsigned)(uintptr_t)(lutB4 + i * 256 + t),
            0, 0);
    }
    asm volatile("s_wait_asynccnt 0" ::: "memory");
    __syncthreads();

    const long imgBase = (long)bc * (Hh * Ww);
    for (int r = 0; r < 16; ++r) {
        const int y = yb + r;
        float ty = fminf(fmaxf((y + 0.5f) / 128.0f - 0.5f, 0.0f), 7.0f);
        const float wy = ty - (float)y0;
        #pragma unroll
        for (int xi = 0; xi < 4; ++xi) {
            const int x   = xi * 256 + t;
            const long idx = imgBase + (long)y * Ww + x;
            if (r == 0) __builtin_prefetch(&img[idx + 8 * Ww], 0, 1);

            float tx = fminf(fmaxf((x + 0.5f) / 128.0f - 0.5f, 0.0f), 7.0f);
            const int x0 = (int)tx;
            const int x1 = min(x0 + 1, 7);
            const float wx = tx - (float)x0;

            float v = img[idx];
            v = fminf(fmaxf(v, 0.0f), 1.0f);
            int bin = (int)(v * 256.0f);
            bin = bin > 255 ? 255 : bin;

            const float v00 = lutA[x0 * 256 + bin];
            const float v01 = lutA[x1 * 256 + bin];
            const float v10 = lutB[x0 * 256 + bin];
            const float v11 = lutB[x1 * 256 + bin];
            const float top = (1.0f - wx) * v00 + wx * v01;
            const float bot = (1.0f - wx) * v10 + wx * v11;
            out[idx] = (1.0f - wy) * top + wy * bot;
        }
    }
}

extern "C" void kernel_launch(void* const* d_in, const int* in_sizes, int n_in,
                              void* d_out, int out_size, void* d_ws, size_t ws_size,
                              hipStream_t stream)
{
    const float* img = (const float*)d_in[0];  // (8,3,1024,1024) f32
    float* out = (float*)d_out;                // same shape
    float* lut = (float*)d_ws;                 // 24*64*256 floats = 1.5 MB

    // Pass 1: 1536 tiles, one block each (8 waves; wave 0 runs the WMMA CDF)
    clahe_lut_kernel<<<dim3(BCN * 8 * 8), dim3(256), 0, stream>>>(img, lut);
    // Pass 2: 64 row-bands x 24 images
    clahe_apply_kernel<<<dim3(Hh / 16, BCN), dim3(256), 0, stream>>>(img, lut, out);
}